// BaseTransform_9921374453908
// MI455X (gfx1250) — compile-verified
//
#include <hip/hip_runtime.h>
#include <hip/hip_bf16.h>

// ---------------------------------------------------------------------------
// BEVFusion-style LSS bev_pool for MI455X (gfx1250, wave32).
//   geometry einsum   -> v_wmma_f32_16x16x4_f32 (16 points / WMMA)
//   feature streaming -> global_load_async_to_lds_b128 double-buffered tiles
//   feature scatter   -> ds_load_b128 + global_atomic_add_f32 (output in L2)
// Memory-bound: ~638MB feature stream; output (41.5MB) lives in 192MB L2.
// ---------------------------------------------------------------------------

typedef float v2f __attribute__((ext_vector_type(2)));
typedef float v8f __attribute__((ext_vector_type(8)));

#define NCAM   6
#define ND     118
#define FH_    32
#define FW_    88
#define NCH    80
#define NPTS   (NCAM * ND * FH_ * FW_)      // 1,993,728
#define PTS_PER_CAM (ND * FH_ * FW_)        // 332,288 (divisible by 128)
#define NX_    360
#define NY_    360
#define NZ_    1
#define GRID2D (NX_ * NY_)                  // 129,600
#define CAM_STRIDE 32                       // floats per camera in ws

#define WAVES_PER_BLOCK 8
#define GROUPS_PER_WAVE 8                   // 16-point groups per wave
#define PTS_PER_WAVE   (GROUPS_PER_WAVE * 16)          // 128
#define PTS_PER_BLOCK  (PTS_PER_WAVE * WAVES_PER_BLOCK) // 1024
#define TILE_BYTES     (16 * NCH * 4)       // 5120 B per 16-point tile

// gfx1250 async global->LDS path (device pass only; host pass falls back)
#if defined(__gfx1250__) && \
    __has_builtin(__builtin_amdgcn_global_load_async_to_lds_b128) && \
    __has_builtin(__builtin_amdgcn_s_wait_asynccnt)
#define USE_ASYNC_LDS 1
#else
#define USE_ASYNC_LDS 0
#endif

// ---------------- 3x3 helpers -------------------------------------------------
__device__ inline void inv3(const float m[9], float o[9]) {
    float a = m[0], b = m[1], c = m[2];
    float d = m[3], e = m[4], f = m[5];
    float g = m[6], h = m[7], i = m[8];
    float A = e * i - f * h, B = c * h - b * i, C = b * f - c * e;
    float D = f * g - d * i, E = a * i - c * g, F = c * d - a * f;
    float G = d * h - e * g, H = b * g - a * h, I = a * e - b * d;
    float r = 1.0f / (a * A + b * D + c * G);
    o[0] = A * r; o[1] = B * r; o[2] = C * r;
    o[3] = D * r; o[4] = E * r; o[5] = F * r;
    o[6] = G * r; o[7] = H * r; o[8] = I * r;
}

__device__ inline void mm3(const float x[9], const float y[9], float o[9]) {
    for (int i = 0; i < 3; ++i)
        for (int j = 0; j < 3; ++j)
            o[i * 3 + j] = x[i * 3 + 0] * y[0 * 3 + j] +
                           x[i * 3 + 1] * y[1 * 3 + j] +
                           x[i * 3 + 2] * y[2 * 3 + j];
}

// ---------------- per-camera transform precompute ----------------------------
// ws layout per camera (stride 32 floats):
//   [0..8]  A1  = inv(post_rots)          [9..11]  b1 = -A1 @ post_trans
//   [12..20] M2 = extra_rots@rots@inv(K)  [21..23] b2 = extra_rots@trans + extra_trans
__global__ void prep_kernel(const float* __restrict__ c2l,
                            const float* __restrict__ intr,
                            const float* __restrict__ ida,
                            const float* __restrict__ bda,
                            float* __restrict__ cam) {
    int n = threadIdx.x;
    if (n >= NCAM) return;
    const float* idaN = ida + n * 16;
    const float* kN   = intr + n * 16;
    const float* cN   = c2l + n * 16;

    float pR[9], pT[3], A1[9], K3[9], Kinv[9], R[9], tr[3], er[9], et[3];
    for (int i = 0; i < 3; ++i) {
        for (int j = 0; j < 3; ++j) {
            pR[i * 3 + j] = idaN[i * 4 + j];
            K3[i * 3 + j] = kN[i * 4 + j];
            R [i * 3 + j] = cN[i * 4 + j];
            er[i * 3 + j] = bda[i * 4 + j];
        }
        pT[i] = idaN[i * 4 + 3];
        tr[i] = cN[i * 4 + 3];
        et[i] = bda[i * 4 + 3];
    }
    inv3(pR, A1);
    inv3(K3, Kinv);
    float comb[9], M2[9];
    mm3(R, Kinv, comb);
    mm3(er, comb, M2);

    float* o = cam + n * CAM_STRIDE;
    for (int i = 0; i < 9; ++i) o[i] = A1[i];
    for (int i = 0; i < 3; ++i)
        o[9 + i] = -(A1[i * 3] * pT[0] + A1[i * 3 + 1] * pT[1] + A1[i * 3 + 2] * pT[2]);
    for (int i = 0; i < 9; ++i) o[12 + i] = M2[i];
    for (int i = 0; i < 3; ++i)
        o[21 + i] = er[i * 3] * tr[0] + er[i * 3 + 1] * tr[1] + er[i * 3 + 2] * tr[2] + et[i];
}

// ---------------- output clear -----------------------------------------------
__global__ void zero_kernel(float4* __restrict__ out4, int n4) {
    int i = blockIdx.x * blockDim.x + threadIdx.x;
    if (i < n4) out4[i] = make_float4(0.f, 0.f, 0.f, 0.f);
}

#if USE_ASYNC_LDS
// builtin signature: (v4i AS(1)* src, v4i AS(3)* dst, imm offset, imm cpol)
typedef int v4i __attribute__((vector_size(16)));
typedef __attribute__((address_space(1))) v4i gv4i;
typedef __attribute__((address_space(3))) v4i sv4i;

// one 16-point feature tile (5120B) = 10 async b128 instructions per wave
__device__ inline void async_tile_load(const float* gsrc, unsigned char* lds_dst, int lane) {
    const unsigned char* g = (const unsigned char*)gsrc;
#pragma unroll
    for (int k = 0; k < 10; ++k) {
        __builtin_amdgcn_global_load_async_to_lds_b128(
            (gv4i*)(g + k * 512 + lane * 16),
            (sv4i*)(lds_dst + k * 512 + lane * 16),
            /*offset=*/0, /*cpol=*/0);
    }
}
#endif

// ---------------- fused geometry (WMMA) + async-staged scatter ----------------
__global__ __launch_bounds__(256)
void geom_scatter_kernel(const float* __restrict__ x,
                         const unsigned char* __restrict__ depth_kept,
                         const float* __restrict__ cam,
                         float* __restrict__ out) {
    __shared__ unsigned char smem[WAVES_PER_BLOCK * 2 * TILE_BYTES];  // 80 KB

    const int lane     = threadIdx.x & 31;
    const int wib      = threadIdx.x >> 5;
    const int gwave    = blockIdx.x * WAVES_PER_BLOCK + wib;
    const int waveBase = gwave * PTS_PER_WAVE;      // first point of this wave
    const int j        = lane & 15;                 // point-in-group / A row
    const int half     = lane >> 4;
    unsigned char* mybuf = smem + wib * (2 * TILE_BYTES);

    // camera is uniform across the wave's 128 points (332288 % 128 == 0)
    const int n = waveBase / PTS_PER_CAM;
    const float* cp = cam + n * CAM_STRIDE;

    // ---- WMMA A operand = [M2 | b2] padded to 16x4 (loop-invariant) ----
    //   VGPR0: lanes0-15 K=0, lanes16-31 K=2 ; VGPR1: lanes0-15 K=1, lanes16-31 K=3
    v2f amat;
    if (j < 3) {
        amat.x = half ? cp[12 + j * 3 + 2] : cp[12 + j * 3 + 0];
        amat.y = half ? cp[21 + j]         : cp[12 + j * 3 + 1];
    } else {
        amat.x = 0.0f; amat.y = 0.0f;
    }

#if USE_ASYNC_LDS
    async_tile_load(x + (size_t)waveBase * NCH, mybuf, lane);   // tile 0 -> buf 0
#endif

    for (int i = 0; i < GROUPS_PER_WAVE; ++i) {
        const int base = waveBase + i * 16;

#if USE_ASYNC_LDS
        if (i + 1 < GROUPS_PER_WAVE)                            // prefetch tile i+1
            async_tile_load(x + (size_t)(base + 16) * NCH,
                            mybuf + ((i + 1) & 1) * TILE_BYTES, lane);
#endif

        // ---- decode point (base+j) -> (depth d, row h, col w) ----
        const int p  = base + j;
        int rem      = p - n * PTS_PER_CAM;
        const int dd = rem / (FH_ * FW_);
        rem         -= dd * (FH_ * FW_);
        const int h  = rem / FW_;
        const int w  = rem - h * FW_;

        const float fx = (float)w * (703.0f / 87.0f);
        const float fy = (float)h * (255.0f / 31.0f);
        const float fz = 1.0f + 0.5f * (float)dd;

        // ---- stage 1 (VALU): q = inv(post_rots) @ f + b1 ----
        const float qx = cp[0] * fx + cp[1] * fy + cp[2] * fz + cp[9];
        const float qy = cp[3] * fx + cp[4] * fy + cp[5] * fz + cp[10];
        const float qz = cp[6] * fx + cp[7] * fy + cp[8] * fz + cp[11];
        const float ux = qx * qz, uy = qy * qz, uz = qz;        // nonlinear lift

        // ---- stage 2: p = M2 @ u + b2 via V_WMMA_F32_16X16X4_F32 ----
        // B (4x16): VGPR0 lanes0-15 row0 (ux), lanes16-31 row2 (uz)
        //           VGPR1 lanes0-15 row1 (uy), lanes16-31 row3 (1)
        v2f bmat;
        bmat.x = half ? uz   : ux;
        bmat.y = half ? 1.0f : uy;

        v8f czero = {};
        v8f dmat = __builtin_amdgcn_wmma_f32_16x16x4_f32(
            false, amat, false, bmat, (short)0, czero, false, false);
        // lanes 0-15: dmat[0]=px, dmat[1]=py, dmat[2]=pz of point base+lane

        // ---- voxelize, branchless (lanes>=16 masked off) ----
        const int cx = (int)floorf((dmat[0] + 54.0f) * (1.0f / 0.3f));
        const int cy = (int)floorf((dmat[1] + 54.0f) * (1.0f / 0.3f));
        const int cz = (int)floorf((dmat[2] + 10.0f) * (1.0f / 20.0f));
        bool kept = ((unsigned)cx < (unsigned)NX_) & ((unsigned)cy < (unsigned)NY_) &
                    ((unsigned)cz < (unsigned)NZ_) & (lane < 16) & (depth_kept[p] != 0);
        const int lin = kept ? (cz * NCH) * GRID2D + cx * NY_ + cy : -1;

#if USE_ASYNC_LDS
        // retire tile i (10 async ops); allow tile i+1 (10 more) to stay in flight
        if (i + 1 < GROUPS_PER_WAVE) __builtin_amdgcn_s_wait_asynccnt(10);
        else                         __builtin_amdgcn_s_wait_asynccnt(0);
        const float4* tile = (const float4*)(mybuf + (i & 1) * TILE_BYTES);
#else
        if (base + 16 < NPTS)
            __builtin_prefetch(x + (size_t)(base + 16) * NCH, 0, 0);
        const float4* tile = (const float4*)x + (size_t)base * (NCH / 4);
#endif

        // ---- scatter: 16 points x 20 float4 = 320 vec4, 10 iters x 32 lanes ----
#pragma unroll
        for (int t = 0; t < 10; ++t) {
            const int v  = t * 32 + lane;        // 0..319
            const int pt = v / 20;               // point-in-group 0..15
            const int g  = v - pt * 20;          // float4 group 0..19
            const int ln = __shfl(lin, pt, 32);  // voxel base for that point
            if (ln >= 0) {
                const float4 f4 = tile[pt * (NCH / 4) + g];
                float* o = out + (size_t)ln + (size_t)(g * 4) * GRID2D;
                unsafeAtomicAdd(o + 0 * (size_t)GRID2D, f4.x);
                unsafeAtomicAdd(o + 1 * (size_t)GRID2D, f4.y);
                unsafeAtomicAdd(o + 2 * (size_t)GRID2D, f4.z);
                unsafeAtomicAdd(o + 3 * (size_t)GRID2D, f4.w);
            }
        }
    }
}

// ---------------------------------------------------------------------------
extern "C" void kernel_launch(void* const* d_in, const int* in_sizes, int n_in,
                              void* d_out, int out_size, void* d_ws, size_t ws_size,
                              hipStream_t stream) {
    const float*         x    = (const float*)d_in[0];
    const unsigned char* kept = (const unsigned char*)d_in[1];  // jnp bool -> 1B
    const float*         c2l  = (const float*)d_in[2];
    const float*         intr = (const float*)d_in[3];
    const float*         ida  = (const float*)d_in[4];
    const float*         bda  = (const float*)d_in[5];
    float*               out  = (float*)d_out;
    float*               cam  = (float*)d_ws;   // 6 * 32 floats = 768 B

    // 1) per-camera fused transforms
    prep_kernel<<<1, 32, 0, stream>>>(c2l, intr, ida, bda, cam);

    // 2) clear output (harness poisons it; atomics need zeros every call)
    const int n4 = out_size / 4;                 // out_size = 10,368,000 (div by 4)
    zero_kernel<<<(n4 + 255) / 256, 256, 0, stream>>>((float4*)out, n4);

    // 3) fused geometry (WMMA) + async-staged feature scatter
    const int blocks = NPTS / PTS_PER_BLOCK;     // 1947 — exact, no tail
    geom_scatter_kernel<<<blocks, 256, 0, stream>>>(x, kept, cam, out);
}